// EquivariantLayerNorm_3874060501247
// MI455X (gfx1250) — compile-verified
//
#include <hip/hip_runtime.h>
#include <math.h>

// EquivariantLayerNorm for x:(N,3,1024), weight:(1024)
// Memory-bound streaming kernel on MI455X (gfx1250):
//  - one wave32 per sample, two samples per wave, double-buffered
//    async global->LDS copies (CDNA5 ASYNCcnt data-mover path)
//  - NT hints on the 384MB one-touch input stream and output stream
//  - wave32 butterfly reductions, closed-form 3x3 inverse matrix sqrt

#define D_DIM 1024
#define ROW_BYTES (D_DIM * 4)          // 4096
#define WAVES_PER_BLOCK 4
#define BLOCK_THREADS (WAVES_PER_BLOCK * 32)
#define SAMPLES_PER_WAVE 2

typedef float v4f __attribute__((ext_vector_type(4)));

__device__ __forceinline__ float wave_sum(float v) {
#pragma unroll
  for (int m = 16; m >= 1; m >>= 1) v += __shfl_xor(v, m, 32);
  return v;
}

__device__ __forceinline__ void cross3(const float a[3], const float b[3], float o[3]) {
  o[0] = a[1] * b[2] - a[2] * b[1];
  o[1] = a[2] * b[0] - a[0] * b[2];
  o[2] = a[0] * b[1] - a[1] * b[0];
}

__device__ __forceinline__ float dot3(const float a[3], const float b[3]) {
  return a[0] * b[0] + a[1] * b[1] + a[2] * b[2];
}

// Eigenvector of symmetric 3x3 for eigenvalue e (largest-norm row cross product).
__device__ __forceinline__ void eigvec3(float a00, float a01, float a02,
                                        float a11, float a12, float a22,
                                        float e, float out[3]) {
  float r0[3] = {a00 - e, a01, a02};
  float r1[3] = {a01, a11 - e, a12};
  float r2[3] = {a02, a12, a22 - e};
  float c0[3], c1[3], c2[3];
  cross3(r0, r1, c0);
  cross3(r0, r2, c1);
  cross3(r1, r2, c2);
  float n0 = dot3(c0, c0), n1 = dot3(c1, c1), n2 = dot3(c2, c2);
  float bx = c0[0], by = c0[1], bz = c0[2], nb = n0;
  if (n1 > nb) { bx = c1[0]; by = c1[1]; bz = c1[2]; nb = n1; }
  if (n2 > nb) { bx = c2[0]; by = c2[1]; bz = c2[2]; nb = n2; }
  if (nb > 1e-28f) {
    float inv = rsqrtf(nb);
    out[0] = bx * inv; out[1] = by * inv; out[2] = bz * inv;
  } else {
    out[0] = 1.0f; out[1] = 0.0f; out[2] = 0.0f;
  }
}

// M = V diag(1/sqrt(s_masked + eps)) V^T for symmetric PSD 3x3.
// Mirrors jnp SVD path: s sorted desc, mask s <= s_max*3*machine_eps -> 0.
__device__ __forceinline__ void symsqrtinv3(float c00, float c01, float c02,
                                            float c11, float c12, float c22,
                                            float M[3][3]) {
  const float EPSV = 1e-5f;
  float q = (c00 + c11 + c22) * (1.0f / 3.0f);
  float b00 = c00 - q, b11 = c11 - q, b22 = c22 - q;
  float p2 = b00 * b00 + b11 * b11 + b22 * b22 +
             2.0f * (c01 * c01 + c02 * c02 + c12 * c12);
  float e0, e1, e2;
  float V0[3], V1[3], V2[3];
  if (p2 < 1e-26f) {
    e0 = e1 = e2 = q;
    V0[0] = 1.f; V0[1] = 0.f; V0[2] = 0.f;
    V1[0] = 0.f; V1[1] = 1.f; V1[2] = 0.f;
    V2[0] = 0.f; V2[1] = 0.f; V2[2] = 1.f;
  } else {
    float p = sqrtf(p2 * (1.0f / 6.0f));
    float ip = 1.0f / p;
    float B00 = b00 * ip, B01 = c01 * ip, B02 = c02 * ip;
    float B11 = b11 * ip, B12 = c12 * ip, B22 = b22 * ip;
    float detB = B00 * (B11 * B22 - B12 * B12)
               - B01 * (B01 * B22 - B12 * B02)
               + B02 * (B01 * B12 - B11 * B02);
    float r = 0.5f * detB;
    r = fminf(1.0f, fmaxf(-1.0f, r));
    float phi = acosf(r) * (1.0f / 3.0f);
    e0 = q + 2.0f * p * cosf(phi);                       // largest
    e2 = q + 2.0f * p * cosf(phi + 2.0943951023931953f); // smallest
    e1 = 3.0f * q - e0 - e2;
    eigvec3(c00, c01, c02, c11, c12, c22, e0, V0);
    eigvec3(c00, c01, c02, c11, c12, c22, e2, V2);
    // Gram-Schmidt V2 against V0 (handles near-degenerate pairs)
    float d = dot3(V2, V0);
    V2[0] -= d * V0[0]; V2[1] -= d * V0[1]; V2[2] -= d * V0[2];
    float nn = dot3(V2, V2);
    if (nn > 1e-24f) {
      float inv = rsqrtf(nn);
      V2[0] *= inv; V2[1] *= inv; V2[2] *= inv;
    } else {
      float t[3];
      if (fabsf(V0[0]) < 0.9f) { t[0] = 1.f; t[1] = 0.f; t[2] = 0.f; }
      else                     { t[0] = 0.f; t[1] = 1.f; t[2] = 0.f; }
      cross3(V0, t, V2);
      float inv = rsqrtf(fmaxf(dot3(V2, V2), 1e-30f));
      V2[0] *= inv; V2[1] *= inv; V2[2] *= inv;
    }
    cross3(V2, V0, V1);
  }
  // rank masking like jnp/torch: s > s_max * n * machine_eps else 0
  float smax = fmaxf(e0, fmaxf(e1, e2));
  float thr = smax * 3.0f * 1.1920929e-7f;
  float s0 = (e0 > thr) ? e0 : 0.0f;
  float s1 = (e1 > thr) ? e1 : 0.0f;
  float s2 = (e2 > thr) ? e2 : 0.0f;
  float w0 = rsqrtf(s0 + EPSV), w1 = rsqrtf(s1 + EPSV), w2 = rsqrtf(s2 + EPSV);
#pragma unroll
  for (int i = 0; i < 3; ++i)
#pragma unroll
    for (int j = 0; j < 3; ++j)
      M[i][j] = w0 * V0[i] * V0[j] + w1 * V1[i] * V1[j] + w2 * V2[i] * V2[j];
}

// Issue 24 async B128 global->LDS copies (one 12KB sample), NT input stream.
__device__ __forceinline__ void issue_async_sample(unsigned lds_base,
                                                   unsigned long long gbase) {
#pragma unroll
  for (int r = 0; r < 3; ++r) {
#pragma unroll
    for (int j = 0; j < 8; ++j) {
      unsigned l = lds_base + (unsigned)(r * ROW_BYTES + j * 512);
      unsigned long long g = gbase + (unsigned long long)(r * ROW_BYTES + j * 512);
      asm volatile("global_load_async_to_lds_b128 %0, %1, off th:TH_LOAD_NT"
                   :: "v"(l), "v"(g) : "memory");
    }
  }
}

// Full per-sample pipeline reading a 3x1024 tile that is already in LDS.
__device__ __forceinline__ void process_sample(const float* __restrict__ lbase,
                                               const v4f* __restrict__ w4,
                                               float* __restrict__ ob, int lane) {
  const v4f* l0 = reinterpret_cast<const v4f*>(lbase);
  const v4f* l1 = reinterpret_cast<const v4f*>(lbase + D_DIM);
  const v4f* l2 = reinterpret_cast<const v4f*>(lbase + 2 * D_DIM);

  // ---- pass 1: raw sums S and second moments Q in one LDS sweep ----
  float S0 = 0.f, S1 = 0.f, S2 = 0.f;
  float Q00 = 0.f, Q01 = 0.f, Q02 = 0.f, Q11 = 0.f, Q12 = 0.f, Q22 = 0.f;
#pragma unroll
  for (int j = 0; j < 8; ++j) {
    int idx = j * 32 + lane;
    v4f a = l0[idx], b = l1[idx], c = l2[idx];
#pragma unroll
    for (int k = 0; k < 4; ++k) {
      float av = a[k], bv = b[k], cv = c[k];
      S0 += av; S1 += bv; S2 += cv;
      Q00 = fmaf(av, av, Q00); Q01 = fmaf(av, bv, Q01);
      Q02 = fmaf(av, cv, Q02); Q11 = fmaf(bv, bv, Q11);
      Q12 = fmaf(bv, cv, Q12); Q22 = fmaf(cv, cv, Q22);
    }
  }
  S0 = wave_sum(S0); S1 = wave_sum(S1); S2 = wave_sum(S2);
  Q00 = wave_sum(Q00); Q01 = wave_sum(Q01); Q02 = wave_sum(Q02);
  Q11 = wave_sum(Q11); Q12 = wave_sum(Q12); Q22 = wave_sum(Q22);

  const float invD = 1.0f / (float)D_DIM;
  const float mu0 = S0 * invD, mu1 = S1 * invD, mu2 = S2 * invD;
  const float EPSV = 1e-5f;
  // covar = xc xc^T / D + eps*diag(1,2,3) ; xc xc^T = Q - D mu mu^T
  float c00 = (Q00 * invD - mu0 * mu0) + EPSV * 1.0f;
  float c01 = Q01 * invD - mu0 * mu1;
  float c02 = Q02 * invD - mu0 * mu2;
  float c11 = (Q11 * invD - mu1 * mu1) + EPSV * 2.0f;
  float c12 = Q12 * invD - mu1 * mu2;
  float c22 = (Q22 * invD - mu2 * mu2) + EPSV * 3.0f;

  float M[3][3];
  symsqrtinv3(c00, c01, c02, c11, c12, c22, M);

  // ---- pass 2: out = (M @ xc) * weight, coalesced B128 NT stores ----
#pragma unroll
  for (int j = 0; j < 8; ++j) {
    int idx = j * 32 + lane;
    v4f a = l0[idx], b = l1[idx], c = l2[idx];
    v4f wv = w4[idx];
    v4f r0, r1, r2;
#pragma unroll
    for (int k = 0; k < 4; ++k) {
      float x0 = a[k] - mu0, x1 = b[k] - mu1, x2 = c[k] - mu2;
      r0[k] = (M[0][0] * x0 + M[0][1] * x1 + M[0][2] * x2) * wv[k];
      r1[k] = (M[1][0] * x0 + M[1][1] * x1 + M[1][2] * x2) * wv[k];
      r2[k] = (M[2][0] * x0 + M[2][1] * x1 + M[2][2] * x2) * wv[k];
    }
    __builtin_nontemporal_store(r0, (v4f*)(ob + 0 * D_DIM + idx * 4));
    __builtin_nontemporal_store(r1, (v4f*)(ob + 1 * D_DIM + idx * 4));
    __builtin_nontemporal_store(r2, (v4f*)(ob + 2 * D_DIM + idx * 4));
  }
}

__global__ __launch_bounds__(BLOCK_THREADS) void eqlnorm_kernel(
    const float* __restrict__ x, const float* __restrict__ weight,
    float* __restrict__ out, int N) {
  // double buffer: 2 samples per wave, 4 waves -> 96 KB of the 320 KB WGP LDS
  __shared__ __align__(16) float lds[WAVES_PER_BLOCK][SAMPLES_PER_WAVE][3][D_DIM];
  const int lane = threadIdx.x & 31;
  const int wid = threadIdx.x >> 5;
  const int n0 = (blockIdx.x * WAVES_PER_BLOCK + wid) * SAMPLES_PER_WAVE;
  if (n0 >= N) return; // wave-uniform
  const bool haveB = (n0 + 1) < N;

  const unsigned lane16 = (unsigned)(lane * 16);
  const unsigned ldsA = (unsigned)(uintptr_t)(&lds[wid][0][0][0]) + lane16;
  const unsigned ldsB = (unsigned)(uintptr_t)(&lds[wid][1][0][0]) + lane16;
  const unsigned long long gA =
      (unsigned long long)(uintptr_t)(x + (size_t)n0 * 3 * D_DIM) + lane16;
  const unsigned long long gB = gA + (unsigned long long)(3 * ROW_BYTES);

  // Fill the async queue: sample A then sample B (completion is in-order).
  issue_async_sample(ldsA, gA);
  if (haveB) issue_async_sample(ldsB, gB);

  const v4f* w4 = reinterpret_cast<const v4f*>(weight);
  float* obA = out + (size_t)n0 * 3 * D_DIM;

  // Wait for A only (<=24 outstanding leaves B's DMA in flight), overlap
  // A's compute + stores with B's fetch.
  asm volatile("s_wait_asynccnt 0x18" ::: "memory");
  process_sample(&lds[wid][0][0][0], w4, obA, lane);

  if (haveB) {
    asm volatile("s_wait_asynccnt 0x0" ::: "memory");
    process_sample(&lds[wid][1][0][0], w4, obA + 3 * D_DIM, lane);
  }
}

extern "C" void kernel_launch(void* const* d_in, const int* in_sizes, int n_in,
                              void* d_out, int out_size, void* d_ws, size_t ws_size,
                              hipStream_t stream) {
  (void)n_in; (void)out_size; (void)d_ws; (void)ws_size;
  const float* x = (const float*)d_in[0];
  const float* w = (const float*)d_in[1];
  float* out = (float*)d_out;
  const int N = in_sizes[0] / (3 * D_DIM); // 32768
  const int samples_per_block = WAVES_PER_BLOCK * SAMPLES_PER_WAVE;
  const int blocks = (N + samples_per_block - 1) / samples_per_block;
  eqlnorm_kernel<<<blocks, BLOCK_THREADS, 0, stream>>>(x, w, out, N);
}